// ClusterLoss_22093311771182
// MI455X (gfx1250) — compile-verified
//
#include <hip/hip_runtime.h>

typedef float v2f __attribute__((ext_vector_type(2)));
typedef float v8f __attribute__((ext_vector_type(8)));

#define N_ROWS 16384
#define N_COLS 4096
#define C4     (N_COLS / 4)              // 1024 float4 per row
#define NEG_FLT_MAX (-3.402823466e+38f)

// ---------------------------------------------------------------------------
// 32-lane sum via V_WMMA_F32_16X16X4_F32 with B == ones:
//   A[m][0] = v_m (lanes 0-15), A[m][2] = v_{m+16} (lanes 16-31), A[m][1]=A[m][3]=0
//   => C[m][n] = v_m + v_{m+16}  (replicated across n)
//   sum of the 8 C VGPRs + xor-16 shuffle = full wave sum in every lane.
// Requires EXEC all ones: call only from fully-converged code.
// ---------------------------------------------------------------------------
__device__ __forceinline__ float wave_sum_wmma(float v) {
  v2f a; a[0] = v;    a[1] = 0.0f;
  v2f b; b[0] = 1.0f; b[1] = 1.0f;       // B[k][n] = 1 for all k,n
  v8f c = {};
  c = __builtin_amdgcn_wmma_f32_16x16x4_f32(
      /*neg_a=*/false, a, /*neg_b=*/false, b,
      /*c_mod=*/(short)0, c, /*reuse_a=*/false, /*reuse_b=*/false);
  float s = ((c[0] + c[1]) + (c[2] + c[3])) + ((c[4] + c[5]) + (c[6] + c[7]));
  s += __shfl_xor(s, 16, 32);
  return s;
}

__device__ __forceinline__ float wave_max(float v) {
#pragma unroll
  for (int off = 16; off > 0; off >>= 1)
    v = fmaxf(v, __shfl_xor(v, off, 32));
  return v;
}

// ---------------------------------------------------------------------------
// Fused kernel: each block owns rows [r0, r0+nrows) x all 4096 columns.
// Per row (data held in registers): feed column accumulators AND compute the
// row's softmax entropy (max -> S = sum e^d, T = sum d e^d -> log S - T/S).
// Single pass over HBM; per-block column partials written once at the end.
// ---------------------------------------------------------------------------
__global__ __launch_bounds__(256) void fused_kernel(
    const float* __restrict__ logits,
    float* __restrict__ row_ent,          // [N_ROWS]
    float* __restrict__ partial,          // [gridDim.x][N_COLS]
    int rows_per_block) {
  const int tid  = threadIdx.x;
  const int wid  = tid >> 5;
  const int lane = tid & 31;
  const int r0   = blockIdx.x * rows_per_block;
  int nrows = N_ROWS - r0;
  if (nrows > rows_per_block) nrows = rows_per_block;

  __shared__ float smax[8];
  __shared__ float sS[8];
  __shared__ float sT[8];

  float4 colacc[4];
#pragma unroll
  for (int j = 0; j < 4; ++j) colacc[j] = make_float4(0.0f, 0.0f, 0.0f, 0.0f);

  for (int r = 0; r < nrows; ++r) {       // uniform bound within block
    const int row = r0 + r;
    const float4* p = (const float4*)logits + (size_t)row * C4;

    // prefetch two rows ahead (speculative: OOB clamped, drops silently)
    int prow = row + 2;
    if (prow > N_ROWS - 1) prow = N_ROWS - 1;
    __builtin_prefetch((const float4*)logits + (size_t)prow * C4 + tid, 0, 0);

    float4 x[4];
#pragma unroll
    for (int j = 0; j < 4; ++j) x[j] = p[tid + j * 256];

    // ---- column accumulation (thread-private, no atomics) ----
#pragma unroll
    for (int j = 0; j < 4; ++j) {
      colacc[j].x += x[j].x; colacc[j].y += x[j].y;
      colacc[j].z += x[j].z; colacc[j].w += x[j].w;
    }

    // ---- row max ----
    float m = NEG_FLT_MAX;
#pragma unroll
    for (int j = 0; j < 4; ++j)
      m = fmaxf(m, fmaxf(fmaxf(x[j].x, x[j].y), fmaxf(x[j].z, x[j].w)));
    m = wave_max(m);
    if (lane == 0) smax[wid] = m;
    __syncthreads();
    float rowmax = smax[0];
#pragma unroll
    for (int i = 1; i < 8; ++i) rowmax = fmaxf(rowmax, smax[i]);

    // ---- S = sum e^d, T = sum d e^d ----
    float S = 0.0f, T = 0.0f;
#pragma unroll
    for (int j = 0; j < 4; ++j) {
      float d0 = x[j].x - rowmax, d1 = x[j].y - rowmax;
      float d2 = x[j].z - rowmax, d3 = x[j].w - rowmax;
      float e0 = __expf(d0), e1 = __expf(d1), e2 = __expf(d2), e3 = __expf(d3);
      S += (e0 + e1) + (e2 + e3);
      T += (d0 * e0 + d1 * e1) + (d2 * e2 + d3 * e3);
    }
    float Sw = wave_sum_wmma(S);          // fully converged: EXEC all ones
    float Tw = wave_sum_wmma(T);
    if (lane == 0) { sS[wid] = Sw; sT[wid] = Tw; }
    __syncthreads();
    if (tid == 0) {
      float St = 0.0f, Tt = 0.0f;
#pragma unroll
      for (int i = 0; i < 8; ++i) { St += sS[i]; Tt += sT[i]; }
      row_ent[row] = __logf(St) - Tt / St;
    }
    // cross-iteration safety: smax/sS writes of iter r+1 are ordered after the
    // two barriers above; tid0's sS read precedes its next barrier arrival.
  }

  // ---- per-block column partials: one 16 KB vector per block ----
#pragma unroll
  for (int j = 0; j < 4; ++j)
    ((float4*)partial)[(size_t)blockIdx.x * C4 + tid + j * 256] = colacc[j];
}

// ---------------------------------------------------------------------------
// Fold kernel: reduce [nblk][4096] block partials to [8][4096].
// grid = (16, 8); one column per thread, coalesced across the block.
// ---------------------------------------------------------------------------
__global__ __launch_bounds__(256) void fold_kernel(
    const float* __restrict__ partial,    // [nblk][N_COLS]
    float* __restrict__ partial2,         // [8][N_COLS]
    int nblk) {
  const int c   = blockIdx.x * 256 + threadIdx.x;   // 0..4095
  const int y   = blockIdx.y;                       // 0..7
  const int per = (nblk + 7) / 8;
  int b0 = y * per;
  int b1 = b0 + per;
  if (b1 > nblk) b1 = nblk;
  float s = 0.0f;
  for (int b = b0; b < b1; ++b) s += partial[(size_t)b * N_COLS + c];
  partial2[(size_t)y * N_COLS + c] = s;
}

// ---------------------------------------------------------------------------
// Finalize (single block): mean_logits from 8 partials (registers, no LDS
// tile), entropy of that vector -> L2; mean of row_ent -> L1.
// ---------------------------------------------------------------------------
__global__ __launch_bounds__(256) void finalize_kernel(
    const float* __restrict__ partial2,   // [8][N_COLS]
    const float* __restrict__ row_ent,    // [N_ROWS]
    float* __restrict__ out) {            // out[0]=L1, out[1]=L2
  const int tid  = threadIdx.x;
  const int wid  = tid >> 5;
  const int lane = tid & 31;

  __shared__ float sred[8];
  __shared__ float sred2[8];
  __shared__ float sent[8];

  // ---- mean logits (16 columns per thread, kept in registers) ----
  const float inv = 1.0f / (float)N_ROWS;
  float4 ml[4];
  float lmax = NEG_FLT_MAX;
#pragma unroll
  for (int j = 0; j < 4; ++j) {
    int c4 = tid + j * 256;
    float4 a = make_float4(0.0f, 0.0f, 0.0f, 0.0f);
    for (int b = 0; b < 8; ++b) {
      float4 v = ((const float4*)partial2)[(size_t)b * C4 + c4];
      a.x += v.x; a.y += v.y; a.z += v.z; a.w += v.w;
    }
    a.x *= inv; a.y *= inv; a.z *= inv; a.w *= inv;
    ml[j] = a;
    lmax = fmaxf(lmax, fmaxf(fmaxf(a.x, a.y), fmaxf(a.z, a.w)));
  }
  lmax = wave_max(lmax);
  if (lane == 0) sred[wid] = lmax;
  __syncthreads();
  float M = sred[0];
#pragma unroll
  for (int i = 1; i < 8; ++i) M = fmaxf(M, sred[i]);
  __syncthreads();                        // sred reused below

  // ---- entropy terms of mean-logits vector ----
  float S = 0.0f, T = 0.0f;
#pragma unroll
  for (int j = 0; j < 4; ++j) {
    float d0 = ml[j].x - M, d1 = ml[j].y - M, d2 = ml[j].z - M, d3 = ml[j].w - M;
    float e0 = __expf(d0), e1 = __expf(d1), e2 = __expf(d2), e3 = __expf(d3);
    S += (e0 + e1) + (e2 + e3);
    T += (d0 * e0 + d1 * e1) + (d2 * e2 + d3 * e3);
  }
  float Sw = wave_sum_wmma(S);
  float Tw = wave_sum_wmma(T);

  // ---- mean of per-row entropies (L1) ----
  float e = 0.0f;
  for (int j = 0; j < 64; ++j) e += row_ent[tid + j * 256];
  float Ew = wave_sum_wmma(e);

  if (lane == 0) { sred[wid] = Sw; sred2[wid] = Tw; sent[wid] = Ew; }
  __syncthreads();
  if (tid == 0) {
    float St = 0.0f, Tt = 0.0f, Et = 0.0f;
#pragma unroll
    for (int i = 0; i < 8; ++i) { St += sred[i]; Tt += sred2[i]; Et += sent[i]; }
    out[0] = Et * inv;                    // L1
    out[1] = -(__logf(St) - Tt / St);     // L2
  }
}

// ---------------------------------------------------------------------------
extern "C" void kernel_launch(void* const* d_in, const int* in_sizes, int n_in,
                              void* d_out, int out_size, void* d_ws, size_t ws_size,
                              hipStream_t stream) {
  (void)in_sizes; (void)n_in; (void)out_size;
  const float* logits = (const float*)d_in[0];
  float* out = (float*)d_out;

  // workspace: [16384] row entropies | [8][4096] fold partials | [B][4096] block partials
  float* row_ent  = (float*)d_ws;
  float* partial2 = row_ent + N_ROWS;
  float* partial  = partial2 + 8 * N_COLS;

  // pick block count B from available workspace (target B=1024 -> 16 rows/block)
  size_t fixed = (size_t)(N_ROWS + 8 * N_COLS) * sizeof(float);
  size_t avail = (ws_size > fixed) ? (ws_size - fixed) : 0;
  long   bmax  = (long)(avail / ((size_t)N_COLS * sizeof(float)));
  if (bmax < 1)    bmax = 1;
  if (bmax > 1024) bmax = 1024;
  int rpb = (N_ROWS + (int)bmax - 1) / (int)bmax;   // rows per block
  int B   = (N_ROWS + rpb - 1) / rpb;               // actual block count

  fused_kernel<<<B, 256, 0, stream>>>(logits, row_ent, partial, rpb);
  dim3 g2(16, 8);
  fold_kernel<<<g2, 256, 0, stream>>>(partial, partial2, B);
  finalize_kernel<<<1, 256, 0, stream>>>(partial2, row_ent, out);
}